// SNRLossOnPreComputedAndWindowedFFT_base_17660905521164
// MI455X (gfx1250) — compile-verified
//
#include <hip/hip_runtime.h>
#include <stdint.h>

// SNR loss on precomputed windowed FFT magnitudes (B=4096 rows, F=4096 bins).
// Bandwidth-bound: 128 MiB read -> ~5.8us floor at 23.3 TB/s.
// CDNA5 path: async global->LDS DMA (ASYNCcnt) stages the outputs row while
// the wave computes the targets argmax; window + total sums then come from LDS.

#define F_DIM 4096
#define BLOCK 256
#define WAVES (BLOCK / 32)
// 16-byte chunks per thread to cover one row: F_DIM*4 bytes / 16 / BLOCK = 4
#define CHUNKS_PER_THREAD ((F_DIM * 4) / 16 / BLOCK)
// float4 elements per thread: F_DIM/4/BLOCK = 4
#define VEC4_PER_THREAD (F_DIM / 4 / BLOCK)

__device__ __forceinline__ void async_wait_zero() {
#if __has_builtin(__builtin_amdgcn_s_wait_asynccnt)
    __builtin_amdgcn_s_wait_asynccnt(0);
#else
    asm volatile("s_wait_asynccnt 0x0" ::: "memory");
#endif
}

__global__ __launch_bounds__(BLOCK)
void snr_row_kernel(const float* __restrict__ outputs,
                    const float* __restrict__ targets,
                    float* __restrict__ row_loss)
{
    __shared__ __align__(16) float s_x[F_DIM];  // staged outputs row (16 KB)
    __shared__ float s_v[WAVES];
    __shared__ int   s_i[WAVES];
    __shared__ float s_t[WAVES];
    __shared__ float s_p[WAVES];
    __shared__ int   s_ref;

    const int tid  = threadIdx.x;
    const int lane = tid & 31;
    const int wid  = tid >> 5;
    const long row = blockIdx.x;

    const float* __restrict__ orow = outputs + row * (long)F_DIM;
    const float* __restrict__ trow = targets + row * (long)F_DIM;

    // ---- 1) CDNA5 async DMA: stage outputs row into LDS (b128 per lane) ----
    {
        const uint32_t lds0 = (uint32_t)(uintptr_t)(&s_x[0]);
#pragma unroll
        for (int k = 0; k < CHUNKS_PER_THREAD; ++k) {
            const int      chunk = tid + k * BLOCK;            // 16-byte chunk id
            const uint64_t gaddr = (uint64_t)(uintptr_t)orow + (uint64_t)chunk * 16ull;
            const uint32_t laddr = lds0 + (uint32_t)chunk * 16u;
            asm volatile("global_load_async_to_lds_b128 %0, %1, off"
                         :: "v"(laddr), "v"(gaddr)
                         : "memory");
        }
    }

    // ---- 2) overlapped: argmax of targets^2 (first-occurrence semantics) ----
    float bv = -1.0f;
    int   bi = 0;
#pragma unroll
    for (int k = 0; k < VEC4_PER_THREAD; ++k) {
        const int j4 = tid + k * BLOCK;
        const float4 t = reinterpret_cast<const float4*>(trow)[j4];
        const int b = j4 * 4;
        float v;
        v = t.x * t.x; if (v > bv) { bv = v; bi = b;     }
        v = t.y * t.y; if (v > bv) { bv = v; bi = b + 1; }
        v = t.z * t.z; if (v > bv) { bv = v; bi = b + 2; }
        v = t.w * t.w; if (v > bv) { bv = v; bi = b + 3; }
    }
    // wave32 argmax reduce (tie -> smaller index, matching jnp.argmax)
#pragma unroll
    for (int off = 16; off > 0; off >>= 1) {
        const float ov = __shfl_xor(bv, off, 32);
        const int   oi = __shfl_xor(bi, off, 32);
        if (ov > bv || (ov == bv && oi < bi)) { bv = ov; bi = oi; }
    }
    if (lane == 0) { s_v[wid] = bv; s_i[wid] = bi; }
    __syncthreads();
    if (tid == 0) {
        float mv = s_v[0];
        int   mi = s_i[0];
        for (int i = 1; i < WAVES; ++i) {
            if (s_v[i] > mv || (s_v[i] == mv && s_i[i] < mi)) { mv = s_v[i]; mi = s_i[i]; }
        }
        s_ref = mi;
    }

    // ---- 3) drain async DMA, publish s_x and s_ref ----
    async_wait_zero();
    __syncthreads();

    const int w     = (int)(0.02 * (double)F_DIM) + 1;  // 82 for F=4096
    const int ref   = s_ref;
    int start = ref - w; if (start < 0) start = 0;
    const int end   = ref + w + 1;                      // exclusive; j < F_DIM anyway

    // ---- 4) total + windowed sums of outputs^2 from LDS ----
    float tot = 0.0f, pul = 0.0f;
#pragma unroll
    for (int k = 0; k < VEC4_PER_THREAD; ++k) {
        const int j4 = tid + k * BLOCK;
        const float4 xv = reinterpret_cast<const float4*>(s_x)[j4];
        const int b = j4 * 4;
        const float x0 = xv.x * xv.x;
        const float x1 = xv.y * xv.y;
        const float x2 = xv.z * xv.z;
        const float x3 = xv.w * xv.w;
        tot += x0 + x1 + x2 + x3;
        if (b     >= start && b     < end) pul += x0;
        if (b + 1 >= start && b + 1 < end) pul += x1;
        if (b + 2 >= start && b + 2 < end) pul += x2;
        if (b + 3 >= start && b + 3 < end) pul += x3;
    }
#pragma unroll
    for (int off = 16; off > 0; off >>= 1) {
        tot += __shfl_xor(tot, off, 32);
        pul += __shfl_xor(pul, off, 32);
    }
    if (lane == 0) { s_t[wid] = tot; s_p[wid] = pul; }
    __syncthreads();
    if (tid == 0) {
        float T = 0.0f, P = 0.0f;
        for (int i = 0; i < WAVES; ++i) { T += s_t[i]; P += s_p[i]; }
        const float other = T - P;
        row_loss[row] = -10.0f * log10f(P / (other + 1e-7f));
    }
}

__global__ __launch_bounds__(BLOCK)
void mean_loss_kernel(const float* __restrict__ row_loss,
                      float* __restrict__ out, int B)
{
    __shared__ float s[WAVES];
    float acc = 0.0f;
    for (int j = threadIdx.x; j < B; j += BLOCK) acc += row_loss[j];
#pragma unroll
    for (int off = 16; off > 0; off >>= 1) acc += __shfl_xor(acc, off, 32);
    if ((threadIdx.x & 31) == 0) s[threadIdx.x >> 5] = acc;
    __syncthreads();
    if (threadIdx.x == 0) {
        float t = 0.0f;
        for (int i = 0; i < WAVES; ++i) t += s[i];
        out[0] = t / (float)B;
    }
}

extern "C" void kernel_launch(void* const* d_in, const int* in_sizes, int n_in,
                              void* d_out, int out_size, void* d_ws, size_t ws_size,
                              hipStream_t stream)
{
    (void)n_in; (void)out_size; (void)ws_size;
    const float* outputs = (const float*)d_in[0];  // [B, F] float32
    const float* targets = (const float*)d_in[1];  // [B, F] float32
    float* row_loss = (float*)d_ws;                // B floats of scratch
    float* out      = (float*)d_out;               // scalar float32

    const int B = in_sizes[0] / F_DIM;

    snr_row_kernel<<<B, BLOCK, 0, stream>>>(outputs, targets, row_loss);
    mean_loss_kernel<<<1, BLOCK, 0, stream>>>(row_loss, out, B);
}